// Model_80109730005327
// MI455X (gfx1250) — compile-verified
//
#include <hip/hip_runtime.h>

// D = A(16x4,f32) * B(4x16,f32) + C(16x16,f32), wave32 WMMA.
typedef float v2f __attribute__((ext_vector_type(2)));
typedef float v8f __attribute__((ext_vector_type(8)));

#define LDSPAD 17

__global__ __launch_bounds__(256)
void fused_rnn_mlp_softmax(const float* __restrict__ x0,
                           const float* __restrict__ xs,
                           const float* __restrict__ rw2,   // rnn_w2 [5][2][2]
                           const float* __restrict__ rb2,   // rnn_b2 [5][2]
                           const float* __restrict__ w1,    // [16][16]
                           const float* __restrict__ b1,    // [16]
                           const float* __restrict__ w2,    // [16][16]
                           const float* __restrict__ b2,    // [16]
                           const float* __restrict__ w3,    // [10][16]
                           const float* __restrict__ b3,    // [10]
                           float* __restrict__ out,         // [B][10]
                           int Btot)
{
    __shared__ float Xt[8][16][LDSPAD];   // 8 waves x (16x16 tile, padded)

    const int tid  = threadIdx.x;
    const int wv   = tid >> 5;
    const int lane = tid & 31;
    const int m    = lane & 15;   // tile row (as A) / column N (as B,C,D)
    const int hf   = lane >> 4;   // lane half

    const int b = blockIdx.x * 128 + wv * 16 + m;  // batch row this lane builds

    // ---------------- Phase 1: build 16x16 input tile (wave-private) --------
    // cols 0..5 = x0[b,0..5]; col 6+2r+o = xs[r,63,b,:] . rw2[r][o][:] + rb2[r][o]
    if (hf == 0) {
        const float* px = x0 + (size_t)b * 6;
        v2f p01 = *(const v2f*)(px + 0);
        v2f p23 = *(const v2f*)(px + 2);
        v2f p45 = *(const v2f*)(px + 4);
        Xt[wv][m][0] = p01.x; Xt[wv][m][1] = p01.y;
        Xt[wv][m][2] = p23.x; Xt[wv][m][3] = p23.y;
        Xt[wv][m][4] = p45.x; Xt[wv][m][5] = p45.y;
        v2f xr = *(const v2f*)(xs + ((size_t)63 * (size_t)Btot + (size_t)b) * 2);
        Xt[wv][m][6] = xr.x * rw2[0] + xr.y * rw2[1] + rb2[0];
        Xt[wv][m][7] = xr.x * rw2[2] + xr.y * rw2[3] + rb2[1];
    } else {
        #pragma unroll
        for (int r = 1; r < 5; ++r) {
            v2f xr = *(const v2f*)(xs + ((size_t)(r * 64 + 63) * (size_t)Btot + (size_t)b) * 2);
            const float* wr = rw2 + r * 4;
            Xt[wv][m][6 + 2 * r]     = xr.x * wr[0] + xr.y * wr[1] + rb2[2 * r];
            Xt[wv][m][6 + 2 * r + 1] = xr.x * wr[2] + xr.y * wr[3] + rb2[2 * r + 1];
        }
    }
    asm volatile("s_wait_dscnt 0x0" ::: "memory");   // wave-private tile: DS in-order per wave

    v2f a[4], bb[4];
    v8f acc;

    // ---------------- Layer 1: relu(X @ w1^T + b1) via 4x WMMA f32 ----------
    #pragma unroll
    for (int kk = 0; kk < 4; ++kk) {
        a[kk].x = Xt[wv][m][kk * 4 + hf * 2 + 0];
        a[kk].y = Xt[wv][m][kk * 4 + hf * 2 + 1];
        bb[kk]  = *(const v2f*)(w1 + m * 16 + hf * 2 + kk * 4);   // B[k][n] = w1[n][k]
    }
    {
        float bias = b1[m];
        #pragma unroll
        for (int i = 0; i < 8; ++i) acc[i] = bias;
    }
    #pragma unroll
    for (int kk = 0; kk < 4; ++kk)
        acc = __builtin_amdgcn_wmma_f32_16x16x4_f32(false, a[kk], false, bb[kk],
                                                    (short)0, acc, false, false);
    #pragma unroll
    for (int v = 0; v < 8; ++v)
        Xt[wv][v + 8 * hf][m] = fmaxf(acc[v], 0.0f);
    asm volatile("s_wait_dscnt 0x0" ::: "memory");

    // ---------------- Layer 2: relu(X @ w2^T + b2) ---------------------------
    #pragma unroll
    for (int kk = 0; kk < 4; ++kk) {
        a[kk].x = Xt[wv][m][kk * 4 + hf * 2 + 0];
        a[kk].y = Xt[wv][m][kk * 4 + hf * 2 + 1];
        bb[kk]  = *(const v2f*)(w2 + m * 16 + hf * 2 + kk * 4);
    }
    {
        float bias = b2[m];
        #pragma unroll
        for (int i = 0; i < 8; ++i) acc[i] = bias;
    }
    #pragma unroll
    for (int kk = 0; kk < 4; ++kk)
        acc = __builtin_amdgcn_wmma_f32_16x16x4_f32(false, a[kk], false, bb[kk],
                                                    (short)0, acc, false, false);
    #pragma unroll
    for (int v = 0; v < 8; ++v)
        Xt[wv][v + 8 * hf][m] = fmaxf(acc[v], 0.0f);
    asm volatile("s_wait_dscnt 0x0" ::: "memory");

    // ---------------- Layer 3: logits = X @ w3^T + b3 (w3 is 10x16, pad N) ---
    #pragma unroll
    for (int kk = 0; kk < 4; ++kk) {
        a[kk].x = Xt[wv][m][kk * 4 + hf * 2 + 0];
        a[kk].y = Xt[wv][m][kk * 4 + hf * 2 + 1];
    }
    if (m < 10) {
        #pragma unroll
        for (int kk = 0; kk < 4; ++kk)
            bb[kk] = *(const v2f*)(w3 + m * 16 + hf * 2 + kk * 4);
        float bias = b3[m];
        #pragma unroll
        for (int i = 0; i < 8; ++i) acc[i] = bias;
    } else {
        #pragma unroll
        for (int kk = 0; kk < 4; ++kk) { bb[kk].x = 0.0f; bb[kk].y = 0.0f; }
        #pragma unroll
        for (int i = 0; i < 8; ++i) acc[i] = 0.0f;
    }
    #pragma unroll
    for (int kk = 0; kk < 4; ++kk)
        acc = __builtin_amdgcn_wmma_f32_16x16x4_f32(false, a[kk], false, bb[kk],
                                                    (short)0, acc, false, false);
    if (m < 10) {
        #pragma unroll
        for (int v = 0; v < 8; ++v)
            Xt[wv][v + 8 * hf][m] = acc[v];
    }
    asm volatile("s_wait_dscnt 0x0" ::: "memory");

    // ---------------- Three grouped softmaxes, one row per low-half lane -----
    if (hf == 0) {
        float l[10];
        #pragma unroll
        for (int j = 0; j < 10; ++j) l[j] = Xt[wv][m][j];
        float p[10];
        {
            float mx = fmaxf(l[0], l[1]);
            float e0 = __expf(l[0] - mx), e1 = __expf(l[1] - mx);
            float inv = 1.0f / (e0 + e1);
            p[0] = e0 * inv; p[1] = e1 * inv;
        }
        {
            float mx = fmaxf(fmaxf(l[2], l[3]), fmaxf(l[4], l[5]));
            float e2 = __expf(l[2] - mx), e3 = __expf(l[3] - mx);
            float e4 = __expf(l[4] - mx), e5 = __expf(l[5] - mx);
            float inv = 1.0f / (e2 + e3 + e4 + e5);
            p[2] = e2 * inv; p[3] = e3 * inv; p[4] = e4 * inv; p[5] = e5 * inv;
        }
        {
            float mx = fmaxf(fmaxf(l[6], l[7]), fmaxf(l[8], l[9]));
            float e6 = __expf(l[6] - mx), e7 = __expf(l[7] - mx);
            float e8 = __expf(l[8] - mx), e9 = __expf(l[9] - mx);
            float inv = 1.0f / (e6 + e7 + e8 + e9);
            p[6] = e6 * inv; p[7] = e7 * inv; p[8] = e8 * inv; p[9] = e9 * inv;
        }
        #pragma unroll
        for (int j = 0; j < 10; ++j) Xt[wv][m][j] = p[j];
    }
    __syncthreads();

    // ---------------- Coalesced block-cooperative store: 128 rows x 10 -------
    const size_t outBase = (size_t)blockIdx.x * 128 * 10;
    for (int i = tid; i < 1280; i += 256) {
        int rl = i / 10;
        int cc = i - rl * 10;
        out[outBase + i] = Xt[rl >> 4][rl & 15][cc];
    }
}

extern "C" void kernel_launch(void* const* d_in, const int* in_sizes, int n_in,
                              void* d_out, int out_size, void* d_ws, size_t ws_size,
                              hipStream_t stream) {
    // setup_inputs order: x0, xs, rnn_w1, rnn_b1, rnn_w2, rnn_b2, w1,b1,w2,b2,w3,b3
    const float* x0  = (const float*)d_in[0];
    const float* xs  = (const float*)d_in[1];
    // d_in[2], d_in[3] (rnn_w1, rnn_b1) are dead code in the reference.
    const float* rw2 = (const float*)d_in[4];
    const float* rb2 = (const float*)d_in[5];
    const float* w1  = (const float*)d_in[6];
    const float* b1  = (const float*)d_in[7];
    const float* w2  = (const float*)d_in[8];
    const float* b2  = (const float*)d_in[9];
    const float* w3  = (const float*)d_in[10];
    const float* b3  = (const float*)d_in[11];
    float* out = (float*)d_out;

    const int B = in_sizes[0] / 6;          // x0 is [B,6] -> B = 131072
    const int blocks = B / 128;             // 128 batch rows per block

    hipLaunchKernelGGL(fused_rnn_mlp_softmax, dim3(blocks), dim3(256), 0, stream,
                       x0, xs, rw2, rb2, w1, b1, w2, b2, w3, b3, out, B);
}